// MlpAttention_45079976739210
// MI455X (gfx1250) — compile-verified
//
#include <hip/hip_runtime.h>
#include <hip/hip_bf16.h>
#include <math.h>

#define T_DIM   500
#define B_DIM   32
#define U_DIM   100
#define IN_DIM  1024
#define E_DIM   512
#define M_DIM   512
#define CELL_DIM 1024
#define NCLASS  5000
#define GATES_DIM 4096
#define BODY_K  2048    // [emb(512) | ctx(1024) | p(512)]
#define OUTROW  1536    // OUT + IN

typedef __attribute__((ext_vector_type(2))) float v2f;
typedef __attribute__((ext_vector_type(8))) float v8f;

__device__ __forceinline__ float sigmoidf_(float x) { return 1.0f / (1.0f + expf(-x)); }

// ---------------------------------------------------------------------------
// Init kernels
// ---------------------------------------------------------------------------
__global__ void zero_f32(float* __restrict__ p, int n) {
    int i = blockIdx.x * blockDim.x + threadIdx.x;
    if (i < n) p[i] = 0.0f;
}

// labels = [sos, att_label[:-1]]; embed = emb_W[label] (masked for label<0)
__global__ void embed_kernel(const int* __restrict__ att_label,
                             const float* __restrict__ emb_W,
                             float* __restrict__ embed) {
    int tid = blockIdx.x * blockDim.x + threadIdx.x;
    if (tid >= U_DIM * B_DIM * E_DIM) return;
    int e = tid & (E_DIM - 1);
    int b = (tid >> 9) & (B_DIM - 1);
    int u = tid >> 14;
    int lab = (u == 0) ? (NCLASS - 2) : att_label[(u - 1) * B_DIM + b];
    embed[tid] = (lab < 0) ? 0.0f : emb_W[(size_t)lab * E_DIM + e];
}

__global__ void maskbias_kernel(const float* __restrict__ rnn_mask,
                                float* __restrict__ mask_bias) {
    int i = blockIdx.x * blockDim.x + threadIdx.x;
    if (i < T_DIM * B_DIM) mask_bias[i] = (rnn_mask[i] - 1.0f) * 1e10f;
}

// ---------------------------------------------------------------------------
// att_h = data[16000,1024] @ Wi2h[512,1024]^T  -> [16000, 512]
// One wave per 16x16 tile; 1000 x 32 tiles; K=1024 via WMMA f32 16x16x4.
// ---------------------------------------------------------------------------
__global__ void atth_gemm_kernel(const float* __restrict__ data,
                                 const float* __restrict__ Wi2h,
                                 float* __restrict__ att_h) {
    const int lane = threadIdx.x & 31;
    const int gw   = blockIdx.x * (blockDim.x >> 5) + (threadIdx.x >> 5); // 0..31999 exact
    const int nt   = gw & 31;
    const int mt   = gw >> 5;
    const int half = lane >> 4;
    const int mn   = lane & 15;

    const float* aRow = data + (size_t)(mt * 16 + mn) * IN_DIM + 2 * half;
    const float* wRow = Wi2h + (size_t)(nt * 16 + mn) * IN_DIM + 2 * half;

    v8f acc = {};
#pragma unroll 8
    for (int k0 = 0; k0 < IN_DIM; k0 += 4) {
        v2f a = *(const v2f*)(aRow + k0);
        v2f w = *(const v2f*)(wRow + k0);
        acc = __builtin_amdgcn_wmma_f32_16x16x4_f32(false, a, false, w, (short)0, acc, false, false);
    }
    float* dCol = att_h + (size_t)(nt * 16 + mn);
#pragma unroll
    for (int r = 0; r < 8; ++r)
        dCol[(size_t)(mt * 16 + r + 8 * half) * M_DIM] = acc[r];
}

// ---------------------------------------------------------------------------
// gates[32,4096] = body[32,2048] @ [Wx|Wp]^T + bias   (2 x 256 tiles = 512 waves)
// ---------------------------------------------------------------------------
__global__ void gates_gemm_kernel(const float* __restrict__ body,
                                  const float* __restrict__ Wx,
                                  const float* __restrict__ Wp,
                                  const float* __restrict__ bias,
                                  float* __restrict__ gates) {
    const int lane = threadIdx.x & 31;
    const int gw   = blockIdx.x * (blockDim.x >> 5) + (threadIdx.x >> 5); // 0..511 exact
    const int mt   = gw & 1;
    const int nt   = gw >> 1;
    const int half = lane >> 4;
    const int mn   = lane & 15;
    const int ncol = nt * 16 + mn;

    const float* aRow  = body + (size_t)(mt * 16 + mn) * BODY_K + 2 * half;
    const float* wxRow = Wx + (size_t)ncol * 1536 + 2 * half;
    const float* wpRow = Wp + (size_t)ncol * 512 + 2 * half;

    const float bv = bias[ncol];
    v8f acc = {bv, bv, bv, bv, bv, bv, bv, bv};

#pragma unroll 4
    for (int k0 = 0; k0 < 1536; k0 += 4) {             // body @ Wx^T segment
        v2f a = *(const v2f*)(aRow + k0);
        v2f w = *(const v2f*)(wxRow + k0);
        acc = __builtin_amdgcn_wmma_f32_16x16x4_f32(false, a, false, w, (short)0, acc, false, false);
    }
#pragma unroll 4
    for (int k0 = 0; k0 < 512; k0 += 4) {              // p @ Wp^T segment
        v2f a = *(const v2f*)(aRow + 1536 + k0);
        v2f w = *(const v2f*)(wpRow + k0);
        acc = __builtin_amdgcn_wmma_f32_16x16x4_f32(false, a, false, w, (short)0, acc, false, false);
    }
    float* dCol = gates + ncol;
#pragma unroll
    for (int r = 0; r < 8; ++r)
        dCol[(size_t)(mt * 16 + r + 8 * half) * GATES_DIM] = acc[r];
}

// ---------------------------------------------------------------------------
// LSTM cell elementwise: c' = clip(sig(f)*c + sig(i)*tanh(g)); h = sig(o)*tanh(c')
// ---------------------------------------------------------------------------
__global__ void cell_kernel(const float* __restrict__ gates,
                            float* __restrict__ c, float* __restrict__ h) {
    int tid = blockIdx.x * blockDim.x + threadIdx.x;
    if (tid >= B_DIM * CELL_DIM) return;
    int b = tid >> 10, j = tid & 1023;
    const float* g = gates + (size_t)b * GATES_DIM;
    float ig = sigmoidf_(g[j]);
    float fg = sigmoidf_(g[1024 + j]);
    float gg = tanhf(g[2048 + j]);
    float og = sigmoidf_(g[3072 + j]);
    float cv = fg * c[tid] + ig * gg;
    cv = fminf(fmaxf(cv, -1.0f), 1.0f);
    c[tid] = cv;
    h[tid] = og * tanhf(cv);
}

// ---------------------------------------------------------------------------
// p = h[32,1024] @ Wproj[512,1024]^T -> writes body p-slot and masked output
// ---------------------------------------------------------------------------
__global__ void proj_gemm_kernel(const float* __restrict__ h,
                                 const float* __restrict__ Wproj,
                                 const float* __restrict__ att_mask,
                                 float* __restrict__ body,
                                 float* __restrict__ out, int u) {
    const int lane = threadIdx.x & 31;
    const int gw   = blockIdx.x * (blockDim.x >> 5) + (threadIdx.x >> 5); // 0..63 exact
    const int mt   = gw & 1;
    const int nt   = gw >> 1;
    const int half = lane >> 4;
    const int mn   = lane & 15;
    const int ncol = nt * 16 + mn;

    const float* aRow = h + (size_t)(mt * 16 + mn) * CELL_DIM + 2 * half;
    const float* wRow = Wproj + (size_t)ncol * CELL_DIM + 2 * half;

    v8f acc = {};
#pragma unroll 8
    for (int k0 = 0; k0 < CELL_DIM; k0 += 4) {
        v2f a = *(const v2f*)(aRow + k0);
        v2f w = *(const v2f*)(wRow + k0);
        acc = __builtin_amdgcn_wmma_f32_16x16x4_f32(false, a, false, w, (short)0, acc, false, false);
    }
#pragma unroll
    for (int r = 0; r < 8; ++r) {
        int brow = mt * 16 + r + 8 * half;
        float val = acc[r];
        body[(size_t)brow * BODY_K + 1536 + ncol] = val;                     // p carry
        out[(size_t)u * (B_DIM * OUTROW) + (size_t)brow * OUTROW + ncol] =
            val * att_mask[u * B_DIM + brow];                                // p * am
    }
}

// ---------------------------------------------------------------------------
// state_s = p[32,512] @ Wp2s[512,512]^T  (p read from body slot)
// ---------------------------------------------------------------------------
__global__ void state_gemm_kernel(const float* __restrict__ body,
                                  const float* __restrict__ Wp2s,
                                  float* __restrict__ state_s) {
    const int lane = threadIdx.x & 31;
    const int gw   = blockIdx.x * (blockDim.x >> 5) + (threadIdx.x >> 5); // 0..63 exact
    const int mt   = gw & 1;
    const int nt   = gw >> 1;
    const int half = lane >> 4;
    const int mn   = lane & 15;
    const int ncol = nt * 16 + mn;

    const float* aRow = body + (size_t)(mt * 16 + mn) * BODY_K + 1536 + 2 * half;
    const float* wRow = Wp2s + (size_t)ncol * M_DIM + 2 * half;

    v8f acc = {};
#pragma unroll 8
    for (int k0 = 0; k0 < M_DIM; k0 += 4) {
        v2f a = *(const v2f*)(aRow + k0);
        v2f w = *(const v2f*)(wRow + k0);
        acc = __builtin_amdgcn_wmma_f32_16x16x4_f32(false, a, false, w, (short)0, acc, false, false);
    }
    float* dCol = state_s + ncol;
#pragma unroll
    for (int r = 0; r < 8; ++r)
        dCol[(size_t)(mt * 16 + r + 8 * half) * M_DIM] = acc[r];
}

// ---------------------------------------------------------------------------
// scalar[t,b] = sigmoid( sum_m v[m]*tanh(state_s[b,m] + att_h[t,b,m]) + bias[t,b] )
// One wave per (t,b): 16000 waves; att_h streams from L2.
// ---------------------------------------------------------------------------
__global__ void score_kernel(const float* __restrict__ att_h,
                             const float* __restrict__ state_s,
                             const float* __restrict__ vvec,
                             const float* __restrict__ mask_bias,
                             float* __restrict__ scalar) {
    const int lane = threadIdx.x & 31;
    const int gw   = blockIdx.x * (blockDim.x >> 5) + (threadIdx.x >> 5); // 0..15999 exact
    const int b    = gw & 31;
    const float* ah = att_h + (size_t)gw * M_DIM;
    const float* st = state_s + (size_t)b * M_DIM;

    float partial = 0.0f;
#pragma unroll
    for (int i = 0; i < 16; ++i) {
        int m = lane + 32 * i;
        partial += vvec[m] * tanhf(st[m] + ah[m]);
    }
#pragma unroll
    for (int off = 16; off > 0; off >>= 1)
        partial += __shfl_xor(partial, off, 32);
    if (lane == 0)
        scalar[gw] = sigmoidf_(partial + mask_bias[gw]);
}

// ---------------------------------------------------------------------------
// ctx[b,i] = sum_t data[t,b,i] * scalar[t,b] -> body ctx-slot + masked output
// ---------------------------------------------------------------------------
__global__ void ctx_kernel(const float* __restrict__ data,
                           const float* __restrict__ scalar,
                           const float* __restrict__ att_mask,
                           float* __restrict__ body,
                           float* __restrict__ out, int u) {
    int tid = blockIdx.x * blockDim.x + threadIdx.x;
    if (tid >= B_DIM * IN_DIM) return;
    int b = tid >> 10, i = tid & 1023;
    const float* dp = data + (size_t)b * IN_DIM + i;
    const float* sp = scalar + b;
    float acc = 0.0f;
#pragma unroll 4
    for (int t = 0; t < T_DIM; ++t)
        acc += dp[(size_t)t * (B_DIM * IN_DIM)] * sp[(size_t)t * B_DIM];
    body[(size_t)b * BODY_K + 512 + i] = acc;                               // ctx carry
    out[(size_t)u * (B_DIM * OUTROW) + (size_t)b * OUTROW + 512 + i] =
        acc * att_mask[u * B_DIM + b];
}

// Copy embed[u] into body emb-slot
__global__ void embcopy_kernel(const float* __restrict__ embed,
                               float* __restrict__ body, int u) {
    int tid = blockIdx.x * blockDim.x + threadIdx.x;
    if (tid >= B_DIM * E_DIM) return;
    int b = tid >> 9, e = tid & 511;
    body[(size_t)b * BODY_K + e] = embed[(size_t)(u * B_DIM + b) * E_DIM + e];
}

// ---------------------------------------------------------------------------
extern "C" void kernel_launch(void* const* d_in, const int* in_sizes, int n_in,
                              void* d_out, int out_size, void* d_ws, size_t ws_size,
                              hipStream_t stream) {
    const float* data      = (const float*)d_in[0];
    const float* att_mask  = (const float*)d_in[1];
    const int*   att_label = (const int*)d_in[2];
    const float* rnn_mask  = (const float*)d_in[3];
    const float* emb_W     = (const float*)d_in[4];
    const float* Wi2h      = (const float*)d_in[5];
    const float* Wp2s      = (const float*)d_in[6];
    const float* w_att_v   = (const float*)d_in[7];
    const float* Wx        = (const float*)d_in[8];
    const float* Wp        = (const float*)d_in[9];
    const float* bias      = (const float*)d_in[10];
    const float* Wproj     = (const float*)d_in[11];
    float* out = (float*)d_out;

    float* ws = (float*)d_ws;
    size_t off = 0;
    auto alloc = [&](size_t n) { float* p = ws + off; off += (n + 63) & ~(size_t)63; return p; };
    float* att_h     = alloc((size_t)T_DIM * B_DIM * M_DIM);   // 32.8 MB
    float* embed     = alloc((size_t)U_DIM * B_DIM * E_DIM);   // 6.6 MB
    float* body      = alloc((size_t)B_DIM * BODY_K);
    float* gates     = alloc((size_t)B_DIM * GATES_DIM);
    float* cbuf      = alloc((size_t)B_DIM * CELL_DIM);
    float* hbuf      = alloc((size_t)B_DIM * CELL_DIM);
    float* state_s   = alloc((size_t)B_DIM * M_DIM);
    float* mask_bias = alloc((size_t)T_DIM * B_DIM);
    float* scalar    = alloc((size_t)T_DIM * B_DIM);

    // ---- one-time setup ----
    zero_f32<<<(B_DIM * BODY_K + 255) / 256, 256, 0, stream>>>(body, B_DIM * BODY_K);
    zero_f32<<<(B_DIM * CELL_DIM + 255) / 256, 256, 0, stream>>>(cbuf, B_DIM * CELL_DIM);
    embed_kernel<<<(U_DIM * B_DIM * E_DIM + 255) / 256, 256, 0, stream>>>(att_label, emb_W, embed);
    maskbias_kernel<<<(T_DIM * B_DIM + 255) / 256, 256, 0, stream>>>(rnn_mask, mask_bias);
    atth_gemm_kernel<<<4000, 256, 0, stream>>>(data, Wi2h, att_h);   // 32000 waves exact

    // ---- sequential decode, 100 steps ----
    for (int u = 0; u < U_DIM; ++u) {
        embcopy_kernel<<<(B_DIM * E_DIM + 255) / 256, 256, 0, stream>>>(embed, body, u);
        gates_gemm_kernel<<<64, 256, 0, stream>>>(body, Wx, Wp, bias, gates);   // 512 waves
        cell_kernel<<<(B_DIM * CELL_DIM + 255) / 256, 256, 0, stream>>>(gates, cbuf, hbuf);
        proj_gemm_kernel<<<8, 256, 0, stream>>>(hbuf, Wproj, att_mask, body, out, u); // 64 waves
        state_gemm_kernel<<<8, 256, 0, stream>>>(body, Wp2s, state_s);                // 64 waves
        score_kernel<<<2000, 256, 0, stream>>>(att_h, state_s, w_att_v, mask_bias, scalar); // 16000 waves
        ctx_kernel<<<(B_DIM * IN_DIM + 255) / 256, 256, 0, stream>>>(data, scalar, att_mask, body, out, u);
    }
}